// GCNTransferLearning_41154376630435
// MI455X (gfx1250) — compile-verified
//
#include <hip/hip_runtime.h>
#include <hip/hip_bf16.h>

typedef float v2f __attribute__((ext_vector_type(2)));
typedef float v8f __attribute__((ext_vector_type(8)));
typedef unsigned int u32x4 __attribute__((ext_vector_type(4)));
typedef int i32x4 __attribute__((ext_vector_type(4)));
typedef int i32x8 __attribute__((ext_vector_type(8)));

#define NNODES 50000
#define HDIM   128
#define FDIM   256
#define CDIM   40

// ---------------- utility kernels ----------------

__global__ __launch_bounds__(256) void zero_f32_kernel(float* __restrict__ p, long long n) {
    long long i = (long long)blockIdx.x * blockDim.x + threadIdx.x;
    if (i < n) p[i] = 0.0f;
}

__global__ __launch_bounds__(256) void degree_kernel(const int* __restrict__ src,
                                                     const int* __restrict__ dst,
                                                     float* __restrict__ degO,
                                                     float* __restrict__ degI, int E) {
    int e = blockIdx.x * blockDim.x + threadIdx.x;
    if (e < E) {
        __hip_atomic_fetch_add(&degO[src[e]], 1.0f, __ATOMIC_RELAXED, __HIP_MEMORY_SCOPE_AGENT);
        __hip_atomic_fetch_add(&degI[dst[e]], 1.0f, __ATOMIC_RELAXED, __HIP_MEMORY_SCOPE_AGENT);
    }
}

__global__ __launch_bounds__(256) void rsqrt_kernel(float* __restrict__ degO,
                                                    float* __restrict__ degI, int n) {
    int i = blockIdx.x * blockDim.x + threadIdx.x;
    if (i < n) {
        degO[i] = rsqrtf(fmaxf(degO[i], 1.0f));
        degI[i] = rsqrtf(fmaxf(degI[i], 1.0f));
    }
}

// ---------------- WMMA GEMM ----------------
// A:[M,KDIM] row-major, B:[KDIM,128] row-major, out:[M,128].
// MODE 0: out = A@B + bias[col]           (projection)
// MODE 1: out = (A@B) * rowscale[row]     (per-layer src-side norm fused)
// Grid: M/16 blocks of 256 threads (8 waves); wave w owns N-tile [16w,16w+16).
// A tile (16 x KDIM, contiguous in memory) staged into LDS by the Tensor Data
// Mover with pad_interval=KDIM dwords / pad_amount=4 dwords, reproducing the
// bank-conflict-free stride LDA = KDIM+4.
template <int KDIM, int MODE>
__global__ __launch_bounds__(256) void gemm_wmma_kernel(const float* __restrict__ A,
                                                        const float* __restrict__ B,
                                                        const float* __restrict__ bias,
                                                        const float* __restrict__ rowscale,
                                                        float* __restrict__ out) {
    constexpr int LDA = KDIM + 4;           // pad: row stride 260/132 dwords -> conflict-free
    __shared__ float As[16 * LDA];

    const int m0  = blockIdx.x * 16;
    const int tid = threadIdx.x;

#if defined(__gfx1250__) && __has_builtin(__builtin_amdgcn_tensor_load_to_lds) && \
    __has_builtin(__builtin_amdgcn_s_wait_tensorcnt)
    // --- TDM path: DMA the contiguous 16xKDIM tile into LDS with padding ---
    if (tid < 32) {                          // one wave issues the descriptor
        const unsigned long long ga = (unsigned long long)(uintptr_t)(A + (size_t)m0 * KDIM);
        const unsigned int lds = (unsigned int)(uintptr_t)(&As[0]);
        constexpr unsigned int ELEMS = 16u * KDIM;           // 1-D tile (contiguous rows)
        constexpr int PADIV = (KDIM == 256) ? 7 : 6;         // pad every KDIM dwords
        u32x4 g0;
        g0.x = 1u;                                           // count=1, user descriptor
        g0.y = lds;                                          // lds_addr (bytes)
        g0.z = (unsigned int)ga;                             // global_addr[31:0]
        g0.w = (unsigned int)((ga >> 32) & 0x01FFFFFFu) | (2u << 30); // addr[56:32] | type=2
        i32x8 g1;
        g1[0] = (2 << 16) | (1 << 20) | (PADIV << 22) | (3 << 25); // 4B elems | pad_en | interval | 4dw pad
        g1[1] = (int)((ELEMS & 0xFFFFu) << 16);              // tensor_dim0[15:0]
        g1[2] = (int)(((ELEMS >> 16) & 0xFFFFu) | (1u << 16)); // tensor_dim0[31:16] | tensor_dim1=1
        g1[3] = (int)(ELEMS << 16);                          // tile_dim0 (16 bits)
        g1[4] = 0;                                           // tile_dim1=0, tile_dim2=0 (unused)
        g1[5] = (int)ELEMS;                                  // tensor_dim0_stride (single row: don't-care)
        g1[6] = 0;
        g1[7] = 0;
        i32x4 g2; g2[0] = 1; g2[1] = 1; g2[2] = 0; g2[3] = 0;        // tensor_dim2=1, tensor_dim3=1
        i32x4 g3; g3[0] = 0; g3[1] = (1 << 16); g3[2] = 0; g3[3] = 0; // tensor_dim4=1, tile_dim4=0
#if __clang_major__ >= 23
        i32x8 g1b = {};
        __builtin_amdgcn_tensor_load_to_lds(g0, g1, g2, g3, g1b, 0);
#else
        __builtin_amdgcn_tensor_load_to_lds(g0, g1, g2, g3, 0);
#endif
        __builtin_amdgcn_s_wait_tensorcnt(0);
    }
#else
    // --- fallback: cooperative, coalesced staging through VGPRs ---
    for (int i = tid; i < 16 * KDIM; i += 256) {
        int r = i / KDIM, c = i % KDIM;
        As[r * LDA + c] = A[(size_t)(m0 + r) * KDIM + c];
    }
#endif
    __syncthreads();

    const int lane = tid & 31;
    const int half = lane >> 4;     // 0: K pair {0,1}, 1: K pair {2,3}
    const int l16  = lane & 15;
    const int n0   = (tid >> 5) * 16;

    v8f acc = {};
    for (int k0 = 0; k0 < KDIM; k0 += 4) {
        v2f a, b;
        // A frag: lane(0..15) = row M, VGPRs hold K=k0+2*half, k0+2*half+1
        a.x = As[l16 * LDA + k0 + 2 * half];
        a.y = As[l16 * LDA + k0 + 2 * half + 1];
        // B frag: lane(0..15) = col N, half-wave selects K pair (mirrors A)
        b.x = B[(size_t)(k0 + 2 * half) * HDIM + n0 + l16];
        b.y = B[(size_t)(k0 + 2 * half + 1) * HDIM + n0 + l16];
        acc = __builtin_amdgcn_wmma_f32_16x16x4_f32(false, a, false, b,
                                                    (short)0, acc, false, false);
    }

    // C/D layout: VGPR j -> M = j + 8*half ; N = l16
    const int col = n0 + l16;
    if (MODE == 0) {
        const float bv = bias[col];
#pragma unroll
        for (int j = 0; j < 8; ++j) {
            int row = m0 + j + 8 * half;
            out[(size_t)row * HDIM + col] = acc[j] + bv;
        }
    } else {
#pragma unroll
        for (int j = 0; j < 8; ++j) {
            int row = m0 + j + 8 * half;
            out[(size_t)row * HDIM + col] = acc[j] * rowscale[row];
        }
    }
}

// ---------------- edge aggregation: m[dst] += hs[src] ----------------
// 32 threads per edge, float4 per thread: fully coalesced 512B gather per row.
__global__ __launch_bounds__(256) void aggregate_kernel(const float* __restrict__ hs,
                                                        const int* __restrict__ src,
                                                        const int* __restrict__ dst,
                                                        float* __restrict__ m, int E) {
    long long t = (long long)blockIdx.x * blockDim.x + threadIdx.x;
    int e = (int)(t >> 5);
    if (e >= E) return;
    int f = (int)(t & 31) << 2;
    int s = src[e], d = dst[e];
    const float4 v = *(const float4*)(hs + (size_t)s * HDIM + f);
    float* base = m + (size_t)d * HDIM + f;
    __hip_atomic_fetch_add(base + 0, v.x, __ATOMIC_RELAXED, __HIP_MEMORY_SCOPE_AGENT);
    __hip_atomic_fetch_add(base + 1, v.y, __ATOMIC_RELAXED, __HIP_MEMORY_SCOPE_AGENT);
    __hip_atomic_fetch_add(base + 2, v.z, __ATOMIC_RELAXED, __HIP_MEMORY_SCOPE_AGENT);
    __hip_atomic_fetch_add(base + 3, v.w, __ATOMIC_RELAXED, __HIP_MEMORY_SCOPE_AGENT);
}

// x[i,f] = relu(m[i,f] * rs_in[i] + b[f])
__global__ __launch_bounds__(256) void finalize_kernel(const float* __restrict__ m,
                                                       const float* __restrict__ rs_in,
                                                       const float* __restrict__ b,
                                                       float* __restrict__ x, long long nh) {
    long long i = (long long)blockIdx.x * blockDim.x + threadIdx.x;
    if (i < nh) {
        int node = (int)(i >> 7);
        int f    = (int)(i & (HDIM - 1));
        float v  = fmaf(m[i], rs_in[node], b[f]);
        x[i] = fmaxf(v, 0.0f);
    }
}

// ---------------- classifier: out = x @ Wc + bc  (N=40) ----------------
__global__ __launch_bounds__(64) void classifier_kernel(const float* __restrict__ x,
                                                        const float* __restrict__ Wc,
                                                        const float* __restrict__ bc,
                                                        float* __restrict__ out) {
    __shared__ float xr[HDIM];
    const int row = blockIdx.x;
    const int t   = threadIdx.x;
    xr[t]      = x[(size_t)row * HDIM + t];
    xr[t + 64] = x[(size_t)row * HDIM + t + 64];
    __syncthreads();
    if (t < CDIM) {
        float s = bc[t];
#pragma unroll 4
        for (int k = 0; k < HDIM; ++k) s = fmaf(xr[k], Wc[k * CDIM + t], s);
        out[(size_t)row * CDIM + t] = s;
    }
}

// ---------------- host launch ----------------

extern "C" void kernel_launch(void* const* d_in, const int* in_sizes, int n_in,
                              void* d_out, int out_size, void* d_ws, size_t ws_size,
                              hipStream_t stream) {
    (void)n_in; (void)out_size; (void)ws_size;
    const float* n_feats = (const float*)d_in[0];
    const int*   src     = (const int*)d_in[1];
    const int*   dst     = (const int*)d_in[2];
    const float* Wp      = (const float*)d_in[3];
    const float* bp      = (const float*)d_in[4];
    const float* W1      = (const float*)d_in[5];
    const float* b1      = (const float*)d_in[6];
    const float* W2      = (const float*)d_in[7];
    const float* b2      = (const float*)d_in[8];
    const float* Wc      = (const float*)d_in[9];
    const float* bc      = (const float*)d_in[10];
    float*       out     = (float*)d_out;

    const int E = in_sizes[1];
    const long long NH = (long long)NNODES * HDIM;

    // workspace carve-out (64-float aligned)
    float* ws = (float*)d_ws;
    size_t off = 0;
    auto carve = [&](size_t nelem) { float* p = ws + off; off += (nelem + 63) & ~(size_t)63; return p; };
    float* rs_out = carve(NNODES);        // deg_out accum -> rsqrt
    float* rs_in  = carve(NNODES);        // deg_in  accum -> rsqrt
    float* x      = carve(NH);
    float* hs     = carve(NH);
    float* m      = carve(NH);

    const int B256 = 256;
    dim3 gN((NNODES + B256 - 1) / B256);
    dim3 gE((E + B256 - 1) / B256);
    dim3 gNH((unsigned)((NH + B256 - 1) / B256));
    dim3 gEdgeAgg((unsigned)(((long long)E * 32 + B256 - 1) / B256));
    dim3 gGemm(NNODES / 16);              // 50000 = 3125 * 16 exactly

    // degrees
    zero_f32_kernel<<<gN, B256, 0, stream>>>(rs_out, NNODES);
    zero_f32_kernel<<<gN, B256, 0, stream>>>(rs_in, NNODES);
    degree_kernel<<<gE, B256, 0, stream>>>(src, dst, rs_out, rs_in, E);
    rsqrt_kernel<<<gN, B256, 0, stream>>>(rs_out, rs_in, NNODES);

    // projection: x = n_feats @ Wp + bp
    gemm_wmma_kernel<FDIM, 0><<<gGemm, B256, 0, stream>>>(n_feats, Wp, bp, nullptr, x);

    // GCN layer 1
    gemm_wmma_kernel<HDIM, 1><<<gGemm, B256, 0, stream>>>(x, W1, nullptr, rs_out, hs);
    zero_f32_kernel<<<gNH, B256, 0, stream>>>(m, NH);
    aggregate_kernel<<<gEdgeAgg, B256, 0, stream>>>(hs, src, dst, m, E);
    finalize_kernel<<<gNH, B256, 0, stream>>>(m, rs_in, b1, x, NH);

    // GCN layer 2
    gemm_wmma_kernel<HDIM, 1><<<gGemm, B256, 0, stream>>>(x, W2, nullptr, rs_out, hs);
    zero_f32_kernel<<<gNH, B256, 0, stream>>>(m, NH);
    aggregate_kernel<<<gEdgeAgg, B256, 0, stream>>>(hs, src, dst, m, E);
    finalize_kernel<<<gNH, B256, 0, stream>>>(m, rs_in, b2, x, NH);

    // classifier
    classifier_kernel<<<dim3(NNODES), 64, 0, stream>>>(x, Wc, bc, out);
}